// GaussianMixtureModel_14319420965165
// MI455X (gfx1250) — compile-verified
//
#include <hip/hip_runtime.h>

// ---------------------------------------------------------------------------
// GMM EM refinement for MI455X (gfx1250, wave32, WMMA f32 16x16x4)
//   features [B=4, C=32, N=65536] f32, labels [B, N] i32
//   K=4 classes x S=4 components, 3 EM iterations, output [B, K, N] f32
// ---------------------------------------------------------------------------

#define B_ 4
#define C_ 32
#define N_ 65536
#define K_ 4
#define S_ 4
#define NCOMP (B_ * K_ * S_)   // 64
#define EM_ITERS 3
#define LOG2PI_F 1.8378770664093453f
#define COV_REG_F 1e-4f
#define EPS_F 1e-6f

#define TILE_N 64              // columns staged per stats stage
#define TPAD (TILE_N + 1)      // +1 float pad -> conflict-free strided LDS reads
#define CPAD (C_ + 1)          // same trick for 32x32 precision matrices

typedef __attribute__((ext_vector_type(2))) float v2f;
typedef __attribute__((ext_vector_type(8))) float v8f;

// ---- workspace layout (float offsets) -------------------------------------
#define WS_R      ((size_t)0)                         // [B,K,S,N] responsibilities
#define WS_W      (WS_R + (size_t)NCOMP * N_)         // [64]
#define WS_SUMX   (WS_W + NCOMP)                      // [64][32]
#define WS_SXX    (WS_SUMX + (size_t)NCOMP * C_)      // [64][32*32]
#define WS_MU     (WS_SXX + (size_t)NCOMP * C_ * C_)  // [64][32]
#define WS_PREC   (WS_MU + (size_t)NCOMP * C_)        // [64][32*32]
#define WS_LOGDET (WS_PREC + (size_t)NCOMP * C_ * C_) // [64]

// CDNA5 full-precision WMMA: D(16x16 f32) = A(16x4 f32) * B(4x16 f32) + C
__device__ __forceinline__ v8f wmma4(v2f a, v2f b, v8f c) {
  return __builtin_amdgcn_wmma_f32_16x16x4_f32(
      /*neg_a=*/false, a, /*neg_b=*/false, b,
      /*c_mod=*/(short)0, c, /*reuse_a=*/false, /*reuse_b=*/false);
}

// ---------------------------------------------------------------------------
__global__ void __launch_bounds__(256)
zero_stats_kernel(float* __restrict__ ws) {
  int i = blockIdx.x * blockDim.x + threadIdx.x;
  int n = NCOMP * (1 + C_ + C_ * C_);   // w + sumx + sxx are contiguous
  if (i < n) ws[WS_W + i] = 0.0f;
}

// ---------------------------------------------------------------------------
// M-step sufficient statistics. grid = (chunks=16, comp=64), block = 128
// (4 waves). Each wave owns one 16x16 tile of the 32x32 sxx accumulator and
// chains v_wmma_f32_16x16x4_f32 over the N chunk; r is folded into A.
// 64 columns staged per barrier set -> 16 WMMAs per wave per stage.
// ---------------------------------------------------------------------------
__global__ void __launch_bounds__(128)
stats_kernel(const float* __restrict__ feat, const int* __restrict__ labels,
             float* __restrict__ ws, int mode) {
  __shared__ float xt[C_][TPAD];   // x tile,   [c][n] (padded row: no bank conflicts)
  __shared__ float at[C_][TPAD];   // r*x tile, [c][n]
  __shared__ float rt[TILE_N];     // r values for the staged columns

  const int comp = blockIdx.y;           // (b,k,s) flattened
  const int b = comp >> 4;
  const int k = (comp >> 2) & 3;
  const int s = comp & 3;
  const int tid = threadIdx.x;
  const int wv = tid >> 5;               // wave 0..3
  const int lane = tid & 31;
  const int crow = wv >> 1;              // output row tile 0..1
  const int dcol = wv & 1;               // output col tile 0..1
  const int m = lane & 15;
  const int hi = lane >> 4;

  const int chunk = N_ / gridDim.x;
  const int nbeg = blockIdx.x * chunk;
  const int nend = nbeg + chunk;

  const float* __restrict__ fb = feat + (size_t)b * C_ * N_;
  const float* __restrict__ rb = ws + WS_R + (size_t)comp * N_;
  const int* __restrict__ lb = labels + (size_t)b * N_;

  v8f acc = {};       // 16x16 f32 WMMA accumulator
  float wacc = 0.0f;
  float sxacc = 0.0f;

  for (int n0 = nbeg; n0 < nend; n0 += TILE_N) {
    __syncthreads();  // protect previous tile from overwrite
    if (tid < TILE_N) {
      int n = n0 + tid;
      float rv = (mode == 0)
                     ? ((lb[n] == k && (n & 3) == s) ? 1.0f : 0.0f)
                     : rb[n];
      rt[tid] = rv;
      wacc += rv;
    }
    for (int idx = tid; idx < C_ * TILE_N; idx += 128) {
      int c = idx >> 6, j = idx & (TILE_N - 1);   // coalesced 64-float runs
      xt[c][j] = fb[(size_t)c * N_ + n0 + j];
    }
    if (n0 + TILE_N < nend)   // speculative prefetch of the next tile row
      __builtin_prefetch(&fb[(size_t)(tid & 31) * N_ + n0 + TILE_N], 0, 1);
    __syncthreads();
    for (int idx = tid; idx < C_ * TILE_N; idx += 128) {
      int c = idx >> 6, j = idx & (TILE_N - 1);
      at[c][j] = xt[c][j] * rt[j];
    }
    __syncthreads();
    if (tid < C_) {  // sumx partials: thread tid owns channel row tid
      float t = 0.0f;
#pragma unroll
      for (int j = 0; j < TILE_N; ++j) t += at[tid][j];
      sxacc += t;
    }
    // 16 WMMA steps (K=4 each) consume the 64 staged columns.
    // A layout: lanes 0-15 hold K{0,1}, lanes 16-31 hold K{2,3} (ISA 7.12.2).
    // B layout: VGPR0 rows K{0|1}, VGPR1 rows K{2|3} striped across lane halves.
#pragma unroll
    for (int kk = 0; kk < TILE_N / 4; ++kk) {
      int n = kk * 4;
      v2f a, bf;
      a.x = at[crow * 16 + m][n + hi * 2];
      a.y = at[crow * 16 + m][n + hi * 2 + 1];
      bf.x = xt[dcol * 16 + m][n + hi];
      bf.y = xt[dcol * 16 + m][n + 2 + hi];
      acc = wmma4(a, bf, acc);
    }
  }

  // global reduction across chunk blocks
  if (tid < TILE_N) atomicAdd(&ws[WS_W + comp], wacc);
  if (tid < C_) atomicAdd(&ws[WS_SUMX + (size_t)comp * C_ + tid], sxacc);
  float* __restrict__ sxx = ws + WS_SXX + (size_t)comp * C_ * C_;
#pragma unroll
  for (int i = 0; i < 8; ++i) {   // C/D layout: VGPR i -> row i (lanes<16) / 8+i
    int c = crow * 16 + hi * 8 + i;
    int d = dcol * 16 + m;
    atomicAdd(&sxx[c * C_ + d], acc[i]);
  }
}

// ---------------------------------------------------------------------------
// Per-component parameter finalization: mu, cov, prec = cov^-1, logdet.
// One wave per component; lane = matrix column; pivot column broadcast via
// __shfl so the Gauss-Jordan sweep has no cross-lane LDS hazards.
// ---------------------------------------------------------------------------
__global__ void __launch_bounds__(32)
finalize_kernel(float* __restrict__ ws) {
  __shared__ float a[C_][C_];
  __shared__ float mi[C_][C_];
  __shared__ float mu_s[C_];

  const int comp = blockIdx.x;
  const int j = threadIdx.x;  // column 0..31
  const float w = ws[WS_W + comp];
  const float invw = 1.0f / (w + EPS_F);
  const float* __restrict__ sumx = ws + WS_SUMX + (size_t)comp * C_;
  const float* __restrict__ sxx = ws + WS_SXX + (size_t)comp * C_ * C_;

  mu_s[j] = sumx[j] * invw;
  __syncthreads();
  for (int c = 0; c < C_; ++c) {
    float cov = sxx[c * C_ + j] * invw - mu_s[c] * mu_s[j];
    if (c == j) cov += COV_REG_F;
    a[c][j] = cov;
    mi[c][j] = (c == j) ? 1.0f : 0.0f;
  }
  __syncthreads();

  float ld = 0.0f;
  for (int p = 0; p < C_; ++p) {
    float piv = a[p][p];            // uniform broadcast read
    ld += __logf(fabsf(piv));       // cov is SPD -> positive pivots
    float ip = 1.0f / piv;
    a[p][j] *= ip;
    mi[p][j] *= ip;
    __syncthreads();
    float apj = a[p][j];
    float mpj = mi[p][j];
    float colA = a[j][p];           // lane j holds row j's pivot-column value
    for (int r = 0; r < C_; ++r) {
      if (r == p) continue;
      float f = __shfl(colA, r, 32);
      a[r][j] -= f * apj;
      mi[r][j] -= f * mpj;
    }
    __syncthreads();
  }

  float* __restrict__ prec = ws + WS_PREC + (size_t)comp * C_ * C_;
  for (int c = 0; c < C_; ++c) prec[c * C_ + j] = mi[c][j];
  ws[WS_MU + (size_t)comp * C_ + j] = mu_s[j];
  if (j == 0) ws[WS_LOGDET + comp] = ld;
}

// ---------------------------------------------------------------------------
// E-step: quad = (x-mu)^T prec (x-mu) via WMMA, then masked softmax over s
// (stores r) or, on the last iteration, logsumexp over s + softmax over k.
// grid = (N/128, B), block = 256 (8 waves, 16 columns each).
// ---------------------------------------------------------------------------
__global__ void __launch_bounds__(256)
estep_kernel(const float* __restrict__ feat, const int* __restrict__ labels,
             float* __restrict__ ws, float* __restrict__ out, int last) {
  __shared__ float prec_s[S_][C_][CPAD];  // padded rows: conflict-free A-frags
  __shared__ float mu_sh[S_][C_];
  __shared__ float ct_sh[S_];
  __shared__ float xt[8][C_][16];         // per-wave x tiles

  const int b = blockIdx.y;
  const int tid = threadIdx.x;
  const int wv = tid >> 5;
  const int lane = tid & 31;
  const int col = lane & 15;
  const int hi = lane >> 4;
  const int n0 = blockIdx.x * 128 + wv * 16;
  const int n = n0 + col;

  const float* __restrict__ fb = feat + (size_t)b * C_ * N_;
  for (int idx = lane; idx < C_ * 16; idx += 32) {
    int c = idx >> 4, j = idx & 15;
    xt[wv][c][j] = fb[(size_t)c * N_ + n0 + j];
  }
  const int lbl = labels[(size_t)b * N_ + n];

  float cll[K_];
  for (int k = 0; k < K_; ++k) {
    __syncthreads();
    const int compbase = (b * K_ + k) * S_;
    for (int idx = tid; idx < S_ * C_ * C_; idx += 256) {
      int s = idx >> 10, rest = idx & 1023;
      prec_s[s][rest >> 5][rest & 31] =
          ws[WS_PREC + (size_t)(compbase + s) * C_ * C_ + rest];
    }
    for (int idx = tid; idx < S_ * C_; idx += 256)
      mu_sh[idx >> 5][idx & 31] = ws[WS_MU + (size_t)compbase * C_ + idx];
    if (tid < S_) {
      float w0 = ws[WS_W + compbase + 0], w1 = ws[WS_W + compbase + 1];
      float w2 = ws[WS_W + compbase + 2], w3 = ws[WS_W + compbase + 3];
      float wsum = w0 + w1 + w2 + w3;
      float wj = ws[WS_W + compbase + tid];
      float logpi = __logf((wj + EPS_F) / (wsum + S_ * EPS_F));
      ct_sh[tid] =
          logpi - 0.5f * (C_ * LOG2PI_F + ws[WS_LOGDET + compbase + tid]);
    }
    __syncthreads();

    float lp[S_];
#pragma unroll
    for (int s = 0; s < S_; ++s) {
      float quad = 0.0f;
#pragma unroll
      for (int rt = 0; rt < 2; ++rt) {
        v8f acc = {};
#pragma unroll
        for (int d0 = 0; d0 < C_; d0 += 4) {   // contraction over channel d
          v2f a, bb;
          a.x = prec_s[s][rt * 16 + col][d0 + hi * 2];
          a.y = prec_s[s][rt * 16 + col][d0 + hi * 2 + 1];
          int dl = d0 + hi, dh = d0 + 2 + hi;
          bb.x = xt[wv][dl][col] - mu_sh[s][dl];
          bb.y = xt[wv][dh][col] - mu_sh[s][dh];
          acc = wmma4(a, bb, acc);             // t[c, n] tile
        }
        float part = 0.0f;
#pragma unroll
        for (int i = 0; i < 8; ++i) {          // z[c,n] * t[c,n] partial dot
          int c = rt * 16 + hi * 8 + i;
          part += (xt[wv][c][col] - mu_sh[s][c]) * acc[i];
        }
        quad += part + __shfl_xor(part, 16, 32);  // fold lane halves
      }
      lp[s] = ct_sh[s] - 0.5f * quad;
    }

    float mx = fmaxf(fmaxf(lp[0], lp[1]), fmaxf(lp[2], lp[3]));
    float e0 = __expf(lp[0] - mx), e1 = __expf(lp[1] - mx);
    float e2 = __expf(lp[2] - mx), e3 = __expf(lp[3] - mx);
    float den = e0 + e1 + e2 + e3;
    if (!last) {
      float maskv = (lbl == k) ? 1.0f : 0.0f;
      if (hi == 0) {
        float inv = maskv / den;
        ws[WS_R + (size_t)(compbase + 0) * N_ + n] = e0 * inv;
        ws[WS_R + (size_t)(compbase + 1) * N_ + n] = e1 * inv;
        ws[WS_R + (size_t)(compbase + 2) * N_ + n] = e2 * inv;
        ws[WS_R + (size_t)(compbase + 3) * N_ + n] = e3 * inv;
      }
    } else {
      cll[k] = mx + __logf(den);   // logsumexp over components
    }
  }

  if (last && hi == 0) {
    float mx = fmaxf(fmaxf(cll[0], cll[1]), fmaxf(cll[2], cll[3]));
    float e0 = __expf(cll[0] - mx), e1 = __expf(cll[1] - mx);
    float e2 = __expf(cll[2] - mx), e3 = __expf(cll[3] - mx);
    float inv = 1.0f / (e0 + e1 + e2 + e3);
    out[((size_t)b * K_ + 0) * N_ + n] = e0 * inv;
    out[((size_t)b * K_ + 1) * N_ + n] = e1 * inv;
    out[((size_t)b * K_ + 2) * N_ + n] = e2 * inv;
    out[((size_t)b * K_ + 3) * N_ + n] = e3 * inv;
  }
}

// ---------------------------------------------------------------------------
extern "C" void kernel_launch(void* const* d_in, const int* in_sizes, int n_in,
                              void* d_out, int out_size, void* d_ws,
                              size_t ws_size, hipStream_t stream) {
  const float* feat = (const float*)d_in[0];
  const int* labels = (const int*)d_in[1];
  float* ws = (float*)d_ws;
  float* out = (float*)d_out;

  const int zn = NCOMP * (1 + C_ + C_ * C_);
  dim3 statsGrid(16, NCOMP);           // 16 N-chunks x 64 components
  dim3 estepGrid(N_ / 128, B_);

  for (int it = 0; it < EM_ITERS; ++it) {
    hipLaunchKernelGGL(zero_stats_kernel, dim3((zn + 255) / 256), dim3(256), 0,
                       stream, ws);
    hipLaunchKernelGGL(stats_kernel, statsGrid, dim3(128), 0, stream, feat,
                       labels, ws, it == 0 ? 0 : 1);
    hipLaunchKernelGGL(finalize_kernel, dim3(NCOMP), dim3(32), 0, stream, ws);
    hipLaunchKernelGGL(estep_kernel, estepGrid, dim3(256), 0, stream, feat,
                       labels, ws, out, it == EM_ITERS - 1 ? 1 : 0);
  }
}